// CustomEmbedding_69707319214556
// MI455X (gfx1250) — compile-verified
//
#include <hip/hip_runtime.h>
#include <stdint.h>

// Embedding gather: out[t, :] = W[:, x[t]] + b   (out: [16*8192, 512] f32)
//
// MI455X strategy (reasoned, not measured):
//  - Output is 256 MB -> HBM-write-bound, floor ~ 256MB / 23.3 TB/s ~= 11 us.
//  - Table is 8 MB -> lives in the 192 MB L2 after first touch; gather reads
//    are effectively L2 reads, not HBM reads.
//  - Dense one-hot WMMA GEMM would be ~550 GFLOP + >=512 MB of one-hot
//    traffic: strictly worse. The CDNA5 feature that fits this op is the
//    Tensor Data Mover's gather mode (CDNA5 ISA):
//    one TENSOR_LOAD_TO_LDS DMA fetches 16 indexed rows (16-bit indices),
//    one TENSOR_STORE_FROM_LDS writes the contiguous 32 KB result block.
//
// Toolchain note: clang-23/therock-10.0 lane -> TDM builtins take 6 args
// (g0 v4u, g1 v8i, g2 v4i, g3 v4i, extra v8i, cpol i32); extra v8i zeroed.
//
// gather_tdm is defined FIRST so the disasm snippet shows the TDM kernel.

#define NUM_CLASSES 4096
#define EMB_DIM     512
#define NTOK        (16 * 8192)
#define TPG         16                  // tokens per TDM gather (16-bit-index max)

typedef __attribute__((ext_vector_type(4))) unsigned int v4u;
typedef __attribute__((ext_vector_type(8))) int          v8i;
typedef __attribute__((ext_vector_type(4))) int          v4i;

// ---------------------------------------------------------------------------
// TDM gather. One wave (32 threads) per group of 16 tokens.
//   load : D# gather_mode=1, 16x(1 row x 512 f32) indexed rows -> 32 KB LDS
//   store: D# 1-D tile, 8192 f32 LDS -> contiguous output block
// The wave never touches the data in VGPRs; TENSORcnt orders load->store.
// 32 KB LDS/block -> up to 10 blocks per WGP (320 KB LDS); memory-level
// parallelism comes from co-resident blocks each holding one DMA in flight.
// ---------------------------------------------------------------------------
__global__ void __launch_bounds__(32)
gather_tdm(const int* __restrict__ x,
           const float* __restrict__ Wt,
           float* __restrict__ out) {
  __shared__ float buf[TPG * EMB_DIM];          // 32 KB per block

  const int t0 = blockIdx.x * TPG;

  // Pack 16 row indices (uniform across the wave: uniform addresses ->
  // scalar loads; values feed SGPR-sourced descriptor groups 2/3).
  uint32_t idx[8];
#pragma unroll
  for (int i = 0; i < 8; ++i) {
    uint32_t lo = (uint32_t)x[t0 + 2 * i]     & 0xFFFFu;
    uint32_t hi = (uint32_t)x[t0 + 2 * i + 1] & 0xFFFFu;
    idx[i] = lo | (hi << 16);
  }

  const uint32_t lds  = (uint32_t)(uintptr_t)(void*)buf;   // LDS byte offset
  const uint64_t src  = (uint64_t)(uintptr_t)Wt;
  const uint64_t dst  = (uint64_t)(uintptr_t)(out + (size_t)t0 * EMB_DIM);

  // ---- D# for gather load: Wt rows[idx[0..15]] -> LDS -----------------
  v4u g0l;
  g0l.x = 1u | (1u << 31);                       // count=1, gather_mode=1, 16b idx
  g0l.y = lds;                                   // lds_addr
  g0l.z = (uint32_t)src;                         // global_addr[31:0]
  g0l.w = ((uint32_t)(src >> 32) & 0x01FFFFFFu)  // global_addr[56:32]
        | (2u << 30);                            // type=2 ("image")

  v8i g1l;
  g1l[0] = (int)(2u << 16);                      // data_size=2 (4 B), mask=0
  g1l[1] = (int)((uint32_t)EMB_DIM << 16);       // tensor_dim0[15:0] = 512
  g1l[2] = (int)((uint32_t)NUM_CLASSES << 16);   // tensor_dim1[15:0] = 4096
  g1l[3] = (int)((uint32_t)EMB_DIM << 16);       // tile_dim0 = 512
  g1l[4] = TPG;                                  // tile_dim1 = #valid indices
  g1l[5] = EMB_DIM;                              // tensor_dim0_stride = 512
  g1l[6] = 0;                                    // dim1_stride ignored (gather)
  g1l[7] = 0;

  v4i g2l, g3l;                                  // row_index_0..15 (16-bit pairs)
  g2l[0] = (int)idx[0]; g2l[1] = (int)idx[1];
  g2l[2] = (int)idx[2]; g2l[3] = (int)idx[3];
  g3l[0] = (int)idx[4]; g3l[1] = (int)idx[5];
  g3l[2] = (int)idx[6]; g3l[3] = (int)idx[7];

  v8i zero8;
  zero8[0] = 0; zero8[1] = 0; zero8[2] = 0; zero8[3] = 0;
  zero8[4] = 0; zero8[5] = 0; zero8[6] = 0; zero8[7] = 0;

  __builtin_amdgcn_tensor_load_to_lds(g0l, g1l, g2l, g3l, zero8, 0);
  __builtin_amdgcn_s_wait_tensorcnt(0);          // LDS tile resident

  // ---- D# for store: contiguous 8192 f32, LDS -> out ------------------
  v4u g0s;
  g0s.x = 1u;                                    // count=1, linear mode
  g0s.y = lds;
  g0s.z = (uint32_t)dst;
  g0s.w = ((uint32_t)(dst >> 32) & 0x01FFFFFFu) | (2u << 30);

  const uint32_t n = TPG * EMB_DIM;              // 8192 elements (fits 16-bit tile)
  v8i g1s;
  g1s[0] = (int)(2u << 16);                      // data_size = 4 B
  g1s[1] = (int)(n << 16);                       // tensor_dim0 = 8192
  g1s[2] = (int)(1u << 16);                      // tensor_dim1 = 1
  g1s[3] = (int)(n << 16);                       // tile_dim0 = 8192
  g1s[4] = 0;                                    // tile_dim1 unused -> 1-D
  g1s[5] = (int)n;                               // tensor_dim0_stride
  g1s[6] = 0;
  g1s[7] = 0;

  v4i z4;
  z4[0] = 0; z4[1] = 0; z4[2] = 0; z4[3] = 0;

  __builtin_amdgcn_tensor_store_from_lds(g0s, g1s, z4, z4, zero8, 0);
  // no explicit wait: S_ENDPGM performs implicit wait-idle (all counters)
}

// ---------------------------------------------------------------------------
// build_table: Wt[c][e] = W[e][c] + b[e]  (8 MB in d_ws), one-time prep.
// One block per class c: lane e writes Wt[c*512+e] -> fully coalesced stores.
// W reads are column-strided but L2-amortized across the 4096 blocks;
// negligible vs the 256 MB gather output.
// ---------------------------------------------------------------------------
__global__ void build_table(const float* __restrict__ W,
                            const float* __restrict__ b,
                            float* __restrict__ Wt) {
  const int c = blockIdx.x;
  for (int e = threadIdx.x; e < EMB_DIM; e += blockDim.x) {
    Wt[(size_t)c * EMB_DIM + e] = W[(size_t)e * NUM_CLASSES + c] + b[e];
  }
}

// ---------------------------------------------------------------------------
extern "C" void kernel_launch(void* const* d_in, const int* in_sizes, int n_in,
                              void* d_out, int out_size, void* d_ws, size_t ws_size,
                              hipStream_t stream) {
  (void)in_sizes; (void)n_in; (void)out_size; (void)ws_size;
  const int*   x  = (const int*)d_in[0];     // [16, 8192] class ids
  const float* W  = (const float*)d_in[1];   // [512, 4096]
  const float* b  = (const float*)d_in[2];   // [512]
  float*       out = (float*)d_out;          // [16, 8192, 512]
  float*       Wt  = (float*)d_ws;           // [4096, 512]  (8 MB scratch)

  build_table<<<NUM_CLASSES, 256, 0, stream>>>(W, b, Wt);
  gather_tdm<<<NTOK / TPG, 32, 0, stream>>>(x, Wt, out);
}